// Classifier_38860864094305
// MI455X (gfx1250) — compile-verified
//
#include <hip/hip_runtime.h>

// Problem constants (match reference)
#define NN 100000
#define EE 3200000
#define BG 64
#define HD 128
#define CC 10

typedef __attribute__((ext_vector_type(2))) float v2f;
typedef __attribute__((ext_vector_type(4))) float v4f;
typedef __attribute__((ext_vector_type(8))) float v8f;

// ---------------- zero workspace ----------------
__global__ void k_zero(float* __restrict__ p, long n) {
    long i = (long)blockIdx.x * blockDim.x + threadIdx.x;
    long stride = (long)gridDim.x * blockDim.x;
    for (; i < n; i += stride) p[i] = 0.0f;
}

// ---------------- degree accumulation ----------------
__global__ void k_degrees(const int* __restrict__ src, const int* __restrict__ dst,
                          float* __restrict__ in_deg, float* __restrict__ out_deg) {
    int e = blockIdx.x * blockDim.x + threadIdx.x;
    if (e < EE) {
        atomicAdd(&out_deg[src[e]], 1.0f);
        atomicAdd(&in_deg[dst[e]], 1.0f);
    }
}

// ---------------- norms ----------------
__global__ void k_norms(const float* __restrict__ in_deg, const float* __restrict__ out_deg,
                        float* __restrict__ nin, float* __restrict__ nout) {
    int i = blockIdx.x * blockDim.x + threadIdx.x;
    if (i < NN) {
        nin[i]  = rsqrtf(fmaxf(in_deg[i], 1.0f));
        nout[i] = rsqrtf(fmaxf(out_deg[i], 1.0f));
    }
}

// ---------------- layer-1 scalar scatter: agg1[dst] += in_deg[src]*norm_out[src] ----------------
__global__ void k_scatter1(const int* __restrict__ src, const int* __restrict__ dst,
                           const float* __restrict__ in_deg, const float* __restrict__ nout,
                           float* __restrict__ agg1) {
    int e = blockIdx.x * blockDim.x + threadIdx.x;
    if (e < EE) {
        int s = src[e];
        atomicAdd(&agg1[dst[e]], in_deg[s] * nout[s]);
    }
}

// ---------------- fused layer-1 dense + layer-2 scatter ----------------
// feat2[src, j] = relu(agg1[src]*nin[src] * W1[j] + b1[j]) * nout[src]   (recomputed per edge)
// agg2[dst, j] += feat2[src, j]
// 32 lanes per edge, 4 columns per lane.
__global__ void k_scatter2(const int* __restrict__ src, const int* __restrict__ dst,
                           const float* __restrict__ agg1, const float* __restrict__ nin,
                           const float* __restrict__ nout,
                           const float* __restrict__ W1, const float* __restrict__ b1,
                           float* __restrict__ agg2) {
    int t = blockIdx.x * blockDim.x + threadIdx.x;
    int e = t >> 5;
    int lane = t & 31;
    if (e >= EE) return;
    int s = src[e];
    int d = dst[e];
    float r  = agg1[s] * nin[s];
    float no = nout[s];
    int j = lane * 4;
    v4f w  = *(const v4f*)(W1 + j);
    v4f bb = *(const v4f*)(b1 + j);
    float* ag = agg2 + (long)d * HD + j;
#pragma unroll
    for (int q = 0; q < 4; ++q) {
        float v = fmaxf(fmaf(r, w[q], bb[q]), 0.0f) * no;
        atomicAdd(&ag[q], v);
    }
}

// ---------------- layer-2 GEMM (WMMA f32 16x16x4) + bias + ReLU + fused graph pooling ----------
// rst2 = (agg2 * nin[:,None]); h2 = relu(rst2 @ W2 + b2); hg_sum[gid[row]] += h2[row]
// One block = one 16-row tile; 8 waves cover 8 column tiles of 16.
__global__ void __launch_bounds__(256) k_gemm_pool(
        const float* __restrict__ agg2, const float* __restrict__ nin,
        const float* __restrict__ W2, const float* __restrict__ b2,
        const int* __restrict__ gid, float* __restrict__ hg_sum) {
    const int wave = threadIdx.x >> 5;        // column tile 0..7
    const int lane = threadIdx.x & 31;
    const int m    = lane & 15;
    const int hi   = lane >> 4;
    const int row0 = blockIdx.x * 16;
    const int col  = wave * 16 + m;           // B/C/D column for this lane
    const int arow = row0 + m;                // A row for this lane
    const float s  = nin[arow];

    // A lane element K indices: k0 + 2*hi, k0 + 2*hi + 1
    const float* Aptr = agg2 + (long)arow * HD + 2 * hi;
    const float* Bptr = W2 + (long)(2 * hi) * HD + col;

    v8f c = {};
#pragma unroll
    for (int k0 = 0; k0 < HD; k0 += 4) {
        v2f a = *(const v2f*)(Aptr + k0);     // 8B-aligned pair
        a.x *= s;
        a.y *= s;
        v2f b;
        b.x = Bptr[(long)k0 * HD];
        b.y = Bptr[(long)k0 * HD + HD];
        c = __builtin_amdgcn_wmma_f32_16x16x4_f32(false, a, false, b,
                                                  (short)0, c, false, false);
    }

    // epilogue: bias + relu; pooled atomics (graph_ids sorted -> endpoint test)
    const float bias = b2[col];
    float vals[8];
#pragma unroll
    for (int v = 0; v < 8; ++v) vals[v] = fmaxf(c[v] + bias, 0.0f);

    int g0  = gid[row0];
    int g15 = gid[row0 + 15];
    if (g0 == g15) {
        float sum = 0.0f;
#pragma unroll
        for (int v = 0; v < 8; ++v) sum += vals[v];
        atomicAdd(&hg_sum[g0 * HD + col], sum);        // 32 atomics / tile
    } else {
#pragma unroll
        for (int v = 0; v < 8; ++v) {                  // rare mixed-graph tile
            int g = gid[row0 + v + 8 * hi];
            atomicAdd(&hg_sum[g * HD + col], vals[v]);
        }
    }
}

// ---------------- per-graph node counts (LDS histogram) ----------------
__global__ void k_counts(const int* __restrict__ gid, float* __restrict__ counts) {
    __shared__ float sc[BG];
    for (int i = threadIdx.x; i < BG; i += blockDim.x) sc[i] = 0.0f;
    __syncthreads();
    int i = blockIdx.x * blockDim.x + threadIdx.x;
    if (i < NN) atomicAdd(&sc[gid[i]], 1.0f);
    __syncthreads();
    for (int i2 = threadIdx.x; i2 < BG; i2 += blockDim.x)
        if (sc[i2] != 0.0f) atomicAdd(&counts[i2], sc[i2]);
}

// ---------------- classifier: out = (hg_sum / counts) @ Wc + bc  (64x10) ----------------
__global__ void k_classify(const float* __restrict__ hg_sum, const float* __restrict__ counts,
                           const float* __restrict__ Wc, const float* __restrict__ bc,
                           float* __restrict__ out) {
    int t = threadIdx.x;
    if (t >= BG * CC) return;
    int b = t / CC;
    int c = t % CC;
    float inv = 1.0f / fmaxf(counts[b], 1.0f);
    float acc = bc[c];
    const float* hrow = hg_sum + b * HD;
#pragma unroll 4
    for (int k = 0; k < HD; ++k)
        acc = fmaf(hrow[k] * inv, Wc[k * CC + c], acc);
    out[t] = acc;
}

extern "C" void kernel_launch(void* const* d_in, const int* in_sizes, int n_in,
                              void* d_out, int out_size, void* d_ws, size_t ws_size,
                              hipStream_t stream) {
    const int*   src = (const int*)d_in[0];
    const int*   dst = (const int*)d_in[1];
    const int*   gid = (const int*)d_in[2];
    const float* W1  = (const float*)d_in[3];
    const float* b1  = (const float*)d_in[4];
    const float* W2  = (const float*)d_in[5];
    const float* b2  = (const float*)d_in[6];
    const float* Wc  = (const float*)d_in[7];
    const float* bc  = (const float*)d_in[8];
    float* out = (float*)d_out;

    // workspace layout (floats)
    float* ws      = (float*)d_ws;
    float* in_deg  = ws;                        // N
    float* out_deg = ws + (long)NN;             // N
    float* nin     = ws + 2L * NN;              // N
    float* nout    = ws + 3L * NN;              // N
    float* agg1    = ws + 4L * NN;              // N
    float* agg2    = ws + 5L * NN;              // N*128
    float* hg_sum  = agg2 + (long)NN * HD;      // B*128
    float* counts  = hg_sum + (long)BG * HD;    // B
    const long total_ws = 5L * NN + (long)NN * HD + (long)BG * HD + BG;

    k_zero    <<<2048, 256, 0, stream>>>(ws, total_ws);
    k_degrees <<<EE / 256, 256, 0, stream>>>(src, dst, in_deg, out_deg);
    k_norms   <<<(NN + 255) / 256, 256, 0, stream>>>(in_deg, out_deg, nin, nout);
    k_scatter1<<<EE / 256, 256, 0, stream>>>(src, dst, in_deg, nout, agg1);
    k_scatter2<<<(EE * 32) / 256, 256, 0, stream>>>(src, dst, agg1, nin, nout, W1, b1, agg2);
    k_gemm_pool<<<NN / 16, 256, 0, stream>>>(agg2, nin, W2, b2, gid, hg_sum);
    k_counts  <<<(NN + 255) / 256, 256, 0, stream>>>(gid, counts);
    k_classify<<<1, 640, 0, stream>>>(hg_sum, counts, Wc, bc, out);
}